// BSRTransform_86182813761697
// MI455X (gfx1250) — compile-verified
//
#include <hip/hip_runtime.h>
#include <hip/hip_bf16.h>

#define NC 20
#define NB 2
#define BB 16
#define CC 3
#define HH 224
#define WW 224
#define THREADS 448                 // 14 wave32 waves; 2 rows in flight
#define PLANE (HH * WW)             // 50176 floats
#define PLANE_BYTES (PLANE * 4)     // 200704 bytes of LDS (<320KB/WGP)
#define CHUNKS (PLANE / 4)          // 12544 x b128 chunks
#define CHUNK_ITERS (CHUNKS / THREADS)  // 28 exact

// One workgroup per (nc, b, c) output plane.
// Stage the 224x224 fp32 source plane into LDS with CDNA5 async global->LDS
// copies, then each thread owns one output column and walks rows doing
// bilinear gathers from LDS with all column geometry hoisted out of the loop.
__global__ __launch_bounds__(THREADS)
void bsr_main_kernel(const float* __restrict__ x,
                     const int*   __restrict__ w_lens,
                     const int*   __restrict__ h_lens,
                     const int*   __restrict__ perm_w,
                     const int*   __restrict__ perm_h,
                     const float* __restrict__ angles,
                     float*       __restrict__ out) {
    extern __shared__ float smem[];   // PLANE floats; dynamic LDS base = 0

    const int blk = blockIdx.x;
    const int nc  = blk / (BB * CC);
    const int rem = blk - nc * (BB * CC);
    const int b   = rem / CC;
    const int c   = rem - b * CC;

    // ---- Stage source plane into LDS (async, uniform trip count) ----
    const float* plane = x + (size_t)(b * CC + c) * PLANE;
    {
        unsigned     lds_off = (unsigned)threadIdx.x * 16u;
        const float* gaddr   = plane + threadIdx.x * 4;
        #pragma unroll
        for (int it = 0; it < CHUNK_ITERS; ++it) {
            asm volatile("global_load_async_to_lds_b128 %0, %1, off"
                         :: "v"(lds_off), "v"(gaddr)
                         : "memory");
            lds_off += THREADS * 16u;
            gaddr   += THREADS * 4;
        }
    }
    asm volatile("s_wait_asynccnt 0x0" ::: "memory");
    __syncthreads();

    // ---- Workgroup-uniform block metadata (NUM_BLOCK == 2) ----
    const int w0  = w_lens[nc * NB + 0];
    const int w1  = w_lens[nc * NB + 1];
    const int h0  = h_lens[nc * NB + 0];
    const int h1  = h_lens[nc * NB + 1];
    const int pw0 = perm_w[nc * NB + 0];
    const int pw1 = perm_w[nc * NB + 1];
    const int ph0 = perm_h[nc * NB + 0];
    const int ph1 = perm_h[nc * NB + 1];

    const int sw0 = pw0 ? w1 : w0;   // out-space width of first permuted col-block
    const int sh0 = ph0 ? h1 : h0;   // out-space height of first permuted row-block

    const float a0  = angles[(nc * NB + 0) * BB + b];
    const float a1  = angles[(nc * NB + 1) * BB + b];
    const float ca0 = __cosf(a0), sa0 = __sinf(a0);
    const float ca1 = __cosf(a1), sa1 = __sinf(a1);

    float* outp = out + (size_t)((nc * BB + b) * CC + c) * PLANE;

    // ---- Per-thread column-invariant geometry ----
    const int i0 = threadIdx.x / WW;        // 0 or 1 (row parity)
    const int j  = threadIdx.x - i0 * WW;   // owned output column

    const int   k     = (j >= sw0) ? 1 : 0;        // output col-block
    const int   wi    = k ? pw1 : pw0;             // source col-block
    const int   loc_j = j - (k ? sw0 : 0);
    const int   Wb    = wi ? w1 : w0;
    const int   sj0   = wi ? w0 : 0;
    const float ca    = k ? ca1 : ca0;
    const float sa    = k ? sa1 : sa0;

    const float cx    = 0.5f * (float)(Wb - 1);
    const float dxv   = (float)loc_j - cx;
    const float xbase = __builtin_fmaf(ca, dxv, cx);   // sx = xbase + sa*dyv
    const float ybase = -sa * dxv;                      // sy = ybase + cy + ca*dyv

    // ---- Row loop: 2 FMAs + bilinear LDS gather per pixel ----
    for (int i = i0; i < HH; i += 2) {
        const int m     = (i >= sh0) ? 1 : 0;     // output row-block
        const int hj    = m ? ph1 : ph0;          // source row-block
        const int loc_i = i - (m ? sh0 : 0);
        const int Hb    = hj ? h1 : h0;
        const int si0   = hj ? h0 : 0;

        const float cy  = 0.5f * (float)(Hb - 1);
        const float dyv = (float)loc_i - cy;

        const float sx = __builtin_fmaf(sa, dyv, xbase);
        const float sy = __builtin_fmaf(ca, dyv, ybase + cy);

        const float flx = floorf(sx);
        const float fly = floorf(sy);
        const int ix0 = (int)flx, iy0 = (int)fly;
        const int ix1 = ix0 + 1,  iy1 = iy0 + 1;
        const float fx = sx - flx;
        const float fy = sy - fly;

        const int cx0 = min(max(ix0, 0), Wb - 1);
        const int cx1 = min(max(ix1, 0), Wb - 1);
        const int cy0 = min(max(iy0, 0), Hb - 1);
        const int cy1 = min(max(iy1, 0), Hb - 1);

        const float vx0 = (ix0 >= 0 && ix0 < Wb) ? 1.0f : 0.0f;
        const float vx1 = (ix1 >= 0 && ix1 < Wb) ? 1.0f : 0.0f;
        const float vy0 = (iy0 >= 0 && iy0 < Hb) ? 1.0f : 0.0f;
        const float vy1 = (iy1 >= 0 && iy1 < Hb) ? 1.0f : 0.0f;

        const int gx0 = sj0 + cx0, gx1 = sj0 + cx1;
        const int r0  = (si0 + cy0) * WW;
        const int r1  = (si0 + cy1) * WW;

        const float s00 = smem[r0 + gx0];
        const float s01 = smem[r0 + gx1];
        const float s10 = smem[r1 + gx0];
        const float s11 = smem[r1 + gx1];

        const float wxa = (1.0f - fx) * vx0;
        const float wxb = fx * vx1;
        const float wya = (1.0f - fy) * vy0;
        const float wyb = fy * vy1;

        const float top = __builtin_fmaf(wxb, s01, wxa * s00);
        const float bot = __builtin_fmaf(wxb, s11, wxa * s10);
        const float acc = __builtin_fmaf(wyb, bot, wya * top);

        outp[i * WW + j] = acc;
    }
}

extern "C" void kernel_launch(void* const* d_in, const int* in_sizes, int n_in,
                              void* d_out, int out_size, void* d_ws, size_t ws_size,
                              hipStream_t stream) {
    (void)in_sizes; (void)n_in; (void)out_size; (void)d_ws; (void)ws_size;
    const float* x      = (const float*)d_in[0];
    const int*   w_lens = (const int*)d_in[1];
    const int*   h_lens = (const int*)d_in[2];
    const int*   perm_w = (const int*)d_in[3];
    const int*   perm_h = (const int*)d_in[4];
    const float* angles = (const float*)d_in[5];
    float*       out    = (float*)d_out;

    const int nblocks = NC * BB * CC;  // 960 workgroups, one output plane each
    bsr_main_kernel<<<nblocks, THREADS, PLANE_BYTES, stream>>>(
        x, w_lens, h_lens, perm_w, perm_h, angles, out);
}